// GatedRetrievalFusion_14920716386854
// MI455X (gfx1250) — compile-verified
//
#include <hip/hip_runtime.h>
#include <math.h>

typedef __attribute__((ext_vector_type(2))) float v2f;
typedef __attribute__((ext_vector_type(4))) float f4;
typedef __attribute__((ext_vector_type(8))) float v8f;

#define DD   768
#define K3   2304
#define HW   50176          // 224*224
#define HW4  12544          // HW/4

// output layout (flat float offsets, reference return order)
#define OFF_PROMPT   0
#define OFF_SEM      768
#define OFF_SEMMAP   1536
#define OFF_SBIAS    38536704
#define OFF_DECMAP   38586880
#define OFF_MLMAP    77122048
#define OFF_ALPHA    77172224
#define OFF_LAM      77172225

// workspace layout (floats)
#define WS_ARAW   0
#define WS_NRAW   768
#define WS_PRAW   1536
#define WS_DRAW   2304
#define WS_SMAP   3072      // alpha*pos[c]
#define WS_DVEC   3840      // dec_vec[c]
#define WS_SCAL   4608      // [alpha, lambda, sb, ml]

// ---------------------------------------------------------------------------
// Kernel 1: four GEMVs y = x @ W via V_WMMA_F32_16X16X4_F32 (exact f32).
// grid = 192 blocks (4 gemvs x 48 column tiles), 128 threads (4 waves).
// A tile: 16x4, row 0 = x[k..k+3], rows 1..15 zero. Waves split K, LDS-reduce.
// ---------------------------------------------------------------------------
__global__ __launch_bounds__(128) void gemv_wmma_f32(
    const float* __restrict__ pos, const float* __restrict__ neg,
    const float* __restrict__ bnd,
    const float* __restrict__ alpha_w1, const float* __restrict__ neg_w1,
    const float* __restrict__ prompt_w, const float* __restrict__ dec_w,
    float* __restrict__ ws)
{
    __shared__ float xf[K3];
    __shared__ float part[4][16];

    const int t = threadIdx.x;
    for (int i = t; i < DD; i += 128) {           // stage fused = [pos|neg|bnd]
        xf[i]          = pos[i];
        xf[i + DD]     = neg[i];
        xf[i + 2 * DD] = bnd[i];
    }
    __syncthreads();

    const int job = blockIdx.x / 48;              // 0:alpha 1:neg 2:prompt 3:dec
    const int n0  = (blockIdx.x % 48) * 16;

    const float* W; const float* x; int K; float* outp;
    if      (job == 0) { W = alpha_w1;  x = xf;      K = K3; outp = ws + WS_ARAW; }
    else if (job == 1) { W = neg_w1;    x = xf + DD; K = DD; outp = ws + WS_NRAW; }
    else if (job == 2) { W = prompt_w;  x = xf;      K = K3; outp = ws + WS_PRAW; }
    else               { W = dec_w;     x = xf;      K = K3; outp = ws + WS_DRAW; }

    const int wave = t >> 5;
    const int lane = t & 31;
    const int qpw  = (K >> 2) >> 2;               // K-quads per wave (4 waves)
    const int k0   = wave * qpw * 4;
    const int kend = k0 + qpw * 4;

    const int n    = lane & 15;
    const int krow = (lane < 16) ? 0 : 2;         // B: VGPR0={K0,K2}, VGPR1={K1,K3}

    v8f c = {};
    for (int k = k0; k < kend; k += 4) {
        v2f a;                                    // A row 0 only (M=0 -> lanes 0,16)
        a.x = (lane == 0) ? x[k]     : ((lane == 16) ? x[k + 2] : 0.0f);
        a.y = (lane == 0) ? x[k + 1] : ((lane == 16) ? x[k + 3] : 0.0f);
        v2f b;
        const float* Wp = W + (k + krow) * DD + n0 + n;
        b.x = Wp[0];
        b.y = Wp[DD];
        // (neg_a, A, neg_b, B, c_mod, C, reuse_a, reuse_b)
        c = __builtin_amdgcn_wmma_f32_16x16x4_f32(false, a, false, b,
                                                  (short)0, c, false, false);
    }
    // D row 0 (M=0) = element 0 of lanes 0..15
    if (lane < 16) part[wave][lane] = c[0];
    __syncthreads();
    if (t < 16)
        outp[n0 + t] = part[0][t] + part[1][t] + part[2][t] + part[3][t];
}

// ---------------------------------------------------------------------------
// Kernel 2: gates (exact-erf GELU + sigmoid), dot reductions, small outputs,
// and per-channel fill values for kernel 3. One block, 256 threads.
// ---------------------------------------------------------------------------
__global__ __launch_bounds__(256) void finalize(
    const float* __restrict__ query, const float* __restrict__ pos,
    const float* __restrict__ neg,   const float* __restrict__ bnd,
    const float* __restrict__ alpha_b1, const float* __restrict__ alpha_w2,
    const float* __restrict__ alpha_b2,
    const float* __restrict__ neg_b1,   const float* __restrict__ neg_w2,
    const float* __restrict__ neg_b2,
    const float* __restrict__ prompt_b, const float* __restrict__ dec_b,
    const float* __restrict__ logit_w,  const float* __restrict__ logit_b,
    float* __restrict__ ws, float* __restrict__ out)
{
    __shared__ float red[5][256];
    const float* a_raw = ws + WS_ARAW;
    const float* n_raw = ws + WS_NRAW;
    const float* p_raw = ws + WS_PRAW;
    const float* d_raw = ws + WS_DRAW;
    float* smap = ws + WS_SMAP;
    float* dvec = ws + WS_DVEC;

    const int t = threadIdx.x;
    const float INV_SQRT2 = 0.70710678118654752f;
    float s0 = 0.f, s1 = 0.f, s2 = 0.f, s3 = 0.f, s4 = 0.f;
    for (int i = t; i < DD; i += 256) {
        float ha = a_raw[i] + alpha_b1[i];
        ha = 0.5f * ha * (1.0f + erff(ha * INV_SQRT2));
        s0 += ha * alpha_w2[i];
        float hn = n_raw[i] + neg_b1[i];
        hn = 0.5f * hn * (1.0f + erff(hn * INV_SQRT2));
        s1 += hn * neg_w2[i];
        float dv = d_raw[i] + dec_b[i];
        dvec[i] = dv;
        float lw = logit_w[i];
        s2 += dv * lw;
        float ng = neg[i];
        s3 += ng * lw;
        float df = bnd[i] - ng;
        s4 += df * df;
    }
    red[0][t] = s0; red[1][t] = s1; red[2][t] = s2; red[3][t] = s3; red[4][t] = s4;
    __syncthreads();
    for (int off = 128; off > 0; off >>= 1) {
        if (t < off)
            for (int q = 0; q < 5; ++q) red[q][t] += red[q][t + off];
        __syncthreads();
    }
    const float alpha = 1.0f / (1.0f + expf(-(red[0][0] + alpha_b2[0])));
    const float lam   = 1.0f / (1.0f + expf(-(red[1][0] + neg_b2[0])));
    const float ldec  = red[2][0] + logit_b[0];
    const float lneg  = red[3][0] + logit_b[0];
    const float ml    = ldec - lam * lneg;
    const float sb    = sqrtf(red[4][0]);

    for (int i = t; i < DD; i += 256) {
        out[OFF_PROMPT + i] = p_raw[i] + prompt_b[i];
        float ap = alpha * pos[i];
        out[OFF_SEM + i] = ap + (1.0f - alpha) * query[i];
        smap[i] = ap;
    }
    if (t == 0) {
        out[OFF_ALPHA] = alpha;
        out[OFF_LAM]   = lam;
        ws[WS_SCAL + 0] = alpha;
        ws[WS_SCAL + 1] = lam;
        ws[WS_SCAL + 2] = sb;
        ws[WS_SCAL + 3] = ml;
    }
}

// ---------------------------------------------------------------------------
// Kernel 3: 309 MB of broadcast fills -> 16B non-temporal streaming stores.
// One block per 224x224 plane: 768 (sem map) + 768 (dec map) + 1 + 1 = 1538.
// ---------------------------------------------------------------------------
__global__ __launch_bounds__(256) void fill_maps(
    const float* __restrict__ ws, float* __restrict__ out)
{
    const int b = blockIdx.x;
    float v; float* base;
    if (b < 768) {
        v = ws[WS_SMAP + b];               base = out + OFF_SEMMAP + (size_t)b * HW;
    } else if (b < 1536) {
        const int c = b - 768;
        v = ws[WS_DVEC + c];               base = out + OFF_DECMAP + (size_t)c * HW;
    } else if (b == 1536) {
        v = ws[WS_SCAL + 2];               base = out + OFF_SBIAS;      // sqrt-norm
    } else {
        v = ws[WS_SCAL + 3];               base = out + OFF_MLMAP;      // mask logit
    }
    f4 val = { v, v, v, v };
    f4* p = (f4*)base;
    for (int j = threadIdx.x; j < HW4; j += 256)
        __builtin_nontemporal_store(val, p + j);
}

// ---------------------------------------------------------------------------
extern "C" void kernel_launch(void* const* d_in, const int* in_sizes, int n_in,
                              void* d_out, int out_size, void* d_ws, size_t ws_size,
                              hipStream_t stream)
{
    const float* query    = (const float*)d_in[0];
    const float* pos      = (const float*)d_in[1];
    const float* neg      = (const float*)d_in[2];
    const float* bnd      = (const float*)d_in[3];
    // d_in[4] = spatial_map: shape-only in the reference, never read.
    const float* alpha_w1 = (const float*)d_in[5];
    const float* alpha_b1 = (const float*)d_in[6];
    const float* alpha_w2 = (const float*)d_in[7];
    const float* alpha_b2 = (const float*)d_in[8];
    const float* neg_w1   = (const float*)d_in[9];
    const float* neg_b1   = (const float*)d_in[10];
    const float* neg_w2   = (const float*)d_in[11];
    const float* neg_b2   = (const float*)d_in[12];
    const float* prompt_w = (const float*)d_in[13];
    const float* prompt_b = (const float*)d_in[14];
    const float* dec_w    = (const float*)d_in[15];
    const float* dec_b    = (const float*)d_in[16];
    const float* logit_w  = (const float*)d_in[17];
    const float* logit_b  = (const float*)d_in[18];

    float* ws  = (float*)d_ws;
    float* out = (float*)d_out;

    gemv_wmma_f32<<<192, 128, 0, stream>>>(pos, neg, bnd,
                                           alpha_w1, neg_w1, prompt_w, dec_w, ws);
    finalize<<<1, 256, 0, stream>>>(query, pos, neg, bnd,
                                    alpha_b1, alpha_w2, alpha_b2,
                                    neg_b1, neg_w2, neg_b2,
                                    prompt_b, dec_b, logit_w, logit_b,
                                    ws, out);
    fill_maps<<<1538, 256, 0, stream>>>(ws, out);
}